// Net_19224273617422
// MI455X (gfx1250) — compile-verified
//
#include <hip/hip_runtime.h>
#include <hip/hip_bf16.h>

typedef __attribute__((ext_vector_type(2))) float v2f;
typedef __attribute__((ext_vector_type(8))) float v8f;

// ---------------------------------------------------------------------------
// degree histogram: deg[dst[e]] += 1
// ---------------------------------------------------------------------------
__global__ void k_deg(const int* __restrict__ ei, float* __restrict__ deg, int E) {
    int e = blockIdx.x * blockDim.x + threadIdx.x;
    if (e < E) atomicAdd(&deg[ei[E + e]], 1.0f);
}

// deg -> deg^-0.5 in place (inf for deg==0, matching torch/jax)
__global__ void k_dinv(float* __restrict__ deg, int N) {
    int i = blockIdx.x * blockDim.x + threadIdx.x;
    if (i < N) deg[i] = 1.0f / sqrtf(deg[i]);
}

// ---------------------------------------------------------------------------
// WMMA node linear 3 -> 16: one wave per 16-node tile, no LDS (rows contiguous)
// ---------------------------------------------------------------------------
__global__ void __launch_bounds__(256) k_lin1_wmma(
    const float* __restrict__ x, const float* __restrict__ w,
    const float* __restrict__ b, float* __restrict__ y, int Nfull) {
    const int wv = threadIdx.x >> 5, ln = threadIdx.x & 31;
    const int base = (blockIdx.x * 8 + wv) * 16;
    if (base >= Nfull) return;                    // uniform per wave
    const int n    = (ln < 16) ? ln : ln - 16;
    const int koff = (ln < 16) ? 0 : 2;
    const int row0 = (ln < 16) ? 0 : 8;

    const float* xr = x + (size_t)(base + n) * 3;
    v2f a, bf;
    a.x  = xr[koff];                              // k = 0 or 2 (always < 3)
    a.y  = (koff + 1 < 3) ? xr[koff + 1] : 0.f;   // k = 1 valid, k = 3 padded
    bf.x = w[koff * 16 + n];
    bf.y = (koff + 1 < 3) ? w[(koff + 1) * 16 + n] : 0.f;

    v8f c; const float bias = b[n];
#pragma unroll
    for (int v = 0; v < 8; ++v) c[v] = bias;
    c = __builtin_amdgcn_wmma_f32_16x16x4_f32(false, a, false, bf, (short)0, c,
                                              false, false);
    float* p = y + (size_t)(base + row0) * 16 + n;
#pragma unroll
    for (int v = 0; v < 8; ++v) p[v * 16] = c[v];
}

__global__ void k_lin1_tail(const float* __restrict__ x, const float* __restrict__ w,
                            const float* __restrict__ b, float* __restrict__ y,
                            int N, int start) {
    int i = start + blockIdx.x * blockDim.x + threadIdx.x;
    if (i >= N) return;
#pragma clang loop unroll(disable)
    for (int o = 0; o < 16; ++o) {
        float s = b[o];
#pragma clang loop unroll(disable)
        for (int k = 0; k < 3; ++k) s = fmaf(x[3 * i + k], w[16 * k + o], s);
        y[16 * i + o] = s;
    }
}

// ---------------------------------------------------------------------------
// WMMA node linear 16 -> 16 with ReLU on input: out = relu(in) @ w + b
// ---------------------------------------------------------------------------
__global__ void __launch_bounds__(256) k_lin16_relu_wmma(
    const float* __restrict__ in, const float* __restrict__ w,
    const float* __restrict__ b, float* __restrict__ out, int Nfull) {
    const int wv = threadIdx.x >> 5, ln = threadIdx.x & 31;
    const int base = (blockIdx.x * 8 + wv) * 16;
    if (base >= Nfull) return;
    const int n    = (ln < 16) ? ln : ln - 16;
    const int koff = (ln < 16) ? 0 : 2;
    const int row0 = (ln < 16) ? 0 : 8;

    const float* r = in + (size_t)(base + n) * 16;
    v8f c; const float bias = b[n];
#pragma unroll
    for (int v = 0; v < 8; ++v) c[v] = bias;
#pragma unroll
    for (int j = 0; j < 4; ++j) {
        const int k0 = j * 4 + koff;
        v2f a, bf;
        const float u0 = r[k0], u1 = r[k0 + 1];
        a.x  = u0 > 0.f ? u0 : 0.f;
        a.y  = u1 > 0.f ? u1 : 0.f;
        bf.x = w[k0 * 16 + n];
        bf.y = w[(k0 + 1) * 16 + n];
        c = __builtin_amdgcn_wmma_f32_16x16x4_f32(false, a, false, bf, (short)0, c,
                                                  false, false);
    }
    float* p = out + (size_t)(base + row0) * 16 + n;
#pragma unroll
    for (int v = 0; v < 8; ++v) p[v * 16] = c[v];
}

__global__ void k_lin16_tail(const float* __restrict__ in, const float* __restrict__ w,
                             const float* __restrict__ b, float* __restrict__ out,
                             int N, int start) {
    int i = start + blockIdx.x * blockDim.x + threadIdx.x;
    if (i >= N) return;
#pragma clang loop unroll(disable)
    for (int o = 0; o < 16; ++o) {
        float s = b[o];
#pragma clang loop unroll(disable)
        for (int k = 0; k < 16; ++k) {
            float u = in[16 * i + k];
            s = fmaf(u > 0.f ? u : 0.f, w[16 * k + o], s);
        }
        out[16 * i + o] = s;
    }
}

// ---------------------------------------------------------------------------
// WMMA final node MLP: out = relu(relu(acc) @ w1 + b1) @ w2 + b2   (16->16->3)
// ---------------------------------------------------------------------------
__global__ void __launch_bounds__(256) k_node_final_wmma(
    const float* __restrict__ acc, const float* __restrict__ w1,
    const float* __restrict__ b1, const float* __restrict__ w2,
    const float* __restrict__ b2, float* __restrict__ out, int Nfull) {
    __shared__ float Ts[8][16 * 17];
    const int wv = threadIdx.x >> 5, ln = threadIdx.x & 31;
    const int base = (blockIdx.x * 8 + wv) * 16;
    if (base >= Nfull) return;
    const int n    = (ln < 16) ? ln : ln - 16;
    const int koff = (ln < 16) ? 0 : 2;
    const int row0 = (ln < 16) ? 0 : 8;
    float* T = Ts[wv];

    const float* r = acc + (size_t)(base + n) * 16;
    v8f h; const float bias = b1[n];
#pragma unroll
    for (int v = 0; v < 8; ++v) h[v] = bias;
#pragma unroll
    for (int j = 0; j < 4; ++j) {
        const int k0 = j * 4 + koff;
        v2f a, bf;
        const float u0 = r[k0], u1 = r[k0 + 1];
        a.x  = u0 > 0.f ? u0 : 0.f;
        a.y  = u1 > 0.f ? u1 : 0.f;
        bf.x = w1[k0 * 16 + n];
        bf.y = w1[(k0 + 1) * 16 + n];
        h = __builtin_amdgcn_wmma_f32_16x16x4_f32(false, a, false, bf, (short)0, h,
                                                  false, false);
    }
#pragma unroll
    for (int v = 0; v < 8; ++v) T[(row0 + v) * 17 + n] = h[v] > 0.f ? h[v] : 0.f;
    __syncthreads();

    // 16 -> 3 tail; lanes<16 do cols {0,1}, lanes>=16 do col {2}
    const int m = n, i = base + m;
    if (ln < 16) {
        float s0 = b2[0], s1 = b2[1];
#pragma unroll
        for (int k = 0; k < 16; ++k) {
            const float t = T[m * 17 + k];
            s0 = fmaf(t, w2[3 * k + 0], s0);
            s1 = fmaf(t, w2[3 * k + 1], s1);
        }
        out[(size_t)i * 3 + 0] = s0;
        out[(size_t)i * 3 + 1] = s1;
    } else {
        float s2 = b2[2];
#pragma unroll
        for (int k = 0; k < 16; ++k) s2 = fmaf(T[m * 17 + k], w2[3 * k + 2], s2);
        out[(size_t)i * 3 + 2] = s2;
    }
}

__global__ void k_node_final_tail(const float* __restrict__ acc, const float* __restrict__ w1,
                                  const float* __restrict__ b1, const float* __restrict__ w2,
                                  const float* __restrict__ b2, float* __restrict__ out,
                                  int N, int start) {
    int i = start + blockIdx.x * blockDim.x + threadIdx.x;
    if (i >= N) return;
    float h[16];
#pragma unroll
    for (int o = 0; o < 16; ++o) {
        float s = b1[o];
#pragma clang loop unroll(disable)
        for (int k = 0; k < 16; ++k) {
            float u = acc[16 * i + k];
            s = fmaf(u > 0.f ? u : 0.f, w1[16 * k + o], s);
        }
        h[o] = s > 0.f ? s : 0.f;
    }
#pragma unroll
    for (int c = 0; c < 3; ++c) {
        float s = b2[c];
#pragma clang loop unroll(disable)
        for (int k = 0; k < 16; ++k) s = fmaf(h[k], w2[3 * k + c], s);
        out[3 * i + c] = s;
    }
}

// ---------------------------------------------------------------------------
// conv1 edge kernel: one wave per FULL 16-edge tile.
//   A[16 x 36] = [ x1[src] | edge_attr | x1[dst] | 0pad ]   (staged in LDS)
//   edge1 = relu(A @ w_e1 + b_e1)        via 9x V_WMMA_F32_16X16X4_F32
//   acc1[dst] += norm * x1[src]          via global_atomic_add_f32
// ---------------------------------------------------------------------------
#define S1   37   // LDS row stride (odd -> fewer bank conflicts)

__global__ void __launch_bounds__(256) k_conv1_edge(
    const float* __restrict__ x1, const float* __restrict__ eattr,
    const int* __restrict__ ei, const float* __restrict__ we,
    const float* __restrict__ be, const float* __restrict__ dis,
    float* __restrict__ edge1, float* __restrict__ acc1, int E, int Efull) {
    __shared__ float As[8][16 * S1];
    __shared__ float nrm[8][16];
    __shared__ int   sdst[8][16];

    const int wv = threadIdx.x >> 5, ln = threadIdx.x & 31;
    const int base = (blockIdx.x * 8 + wv) * 16;
    if (base >= Efull) return;                    // uniform per wave
    float* A = As[wv];

    __builtin_prefetch(ei + base + 2048, 0, 1);
    __builtin_prefetch(ei + E + base + 2048, 0, 1);
    __builtin_prefetch(eattr + (size_t)(base + 2048) * 3, 0, 1);

    // ---- stage A tile (tile is complete: no guards) ----
    if (ln < 16) {
        const int m = ln, e = base + m;
        const int s = ei[e], d = ei[E + e];
        sdst[wv][m] = d;
        nrm[wv][m]  = dis[s] * dis[d];
        const float* xs = x1 + (size_t)s * 16;
#pragma unroll
        for (int f = 0; f < 16; ++f) A[m * S1 + f] = xs[f];
#pragma unroll
        for (int c = 0; c < 3; ++c) A[m * S1 + 16 + c] = eattr[(size_t)e * 3 + c];
        A[m * S1 + 35] = 0.f;                     // K padding
    } else {
        const int m = ln - 16, e = base + m;
        const int d = ei[E + e];
        const float* xd = x1 + (size_t)d * 16;
#pragma unroll
        for (int f = 0; f < 16; ++f) A[m * S1 + 19 + f] = xd[f];
    }
    __syncthreads();

    const int n    = (ln < 16) ? ln : ln - 16;
    const int koff = (ln < 16) ? 0 : 2;
    const int row0 = (ln < 16) ? 0 : 8;

    v2f bf[9];
#pragma unroll
    for (int j = 0; j < 9; ++j) {
        const int k0 = j * 4 + koff;
        bf[j].x = (k0 < 35)     ? we[k0 * 16 + n]       : 0.f;
        bf[j].y = (k0 + 1 < 35) ? we[(k0 + 1) * 16 + n] : 0.f;
    }

    v8f c; const float bias = be[n];
#pragma unroll
    for (int v = 0; v < 8; ++v) c[v] = bias;
#pragma unroll
    for (int j = 0; j < 9; ++j) {
        v2f a;
        a.x = A[n * S1 + j * 4 + koff];
        a.y = A[n * S1 + j * 4 + koff + 1];
        c = __builtin_amdgcn_wmma_f32_16x16x4_f32(false, a, false, bf[j], (short)0, c,
                                                  false, false);
    }

    float* p = edge1 + (size_t)(base + row0) * 16 + n;
#pragma unroll
    for (int v = 0; v < 8; ++v) p[v * 16] = c[v] > 0.f ? c[v] : 0.f;

#pragma unroll
    for (int mm = 0; mm < 8; ++mm) {
        const int m = row0 + mm;
        atomicAdd(&acc1[(size_t)sdst[wv][m] * 16 + n], nrm[wv][m] * A[m * S1 + n]);
    }
}

// compact scalar tail for (E % 16) edges — not launched when E % 16 == 0
__global__ void k_conv1_tail(const float* __restrict__ x1, const float* __restrict__ eattr,
                             const int* __restrict__ ei, const float* __restrict__ we,
                             const float* __restrict__ be, const float* __restrict__ dis,
                             float* __restrict__ edge1, float* __restrict__ acc1,
                             int E, int start) {
    int e = start + blockIdx.x * blockDim.x + threadIdx.x;
    if (e >= E) return;
    const int s = ei[e], d = ei[E + e];
    const float nv = dis[s] * dis[d];
#pragma clang loop unroll(disable)
    for (int o = 0; o < 16; ++o) {
        float acc = be[o];
#pragma clang loop unroll(disable)
        for (int k = 0; k < 16; ++k) acc = fmaf(x1[16 * s + k], we[16 * k + o], acc);
#pragma clang loop unroll(disable)
        for (int k = 0; k < 3; ++k)  acc = fmaf(eattr[(size_t)e * 3 + k], we[16 * (16 + k) + o], acc);
#pragma clang loop unroll(disable)
        for (int k = 0; k < 16; ++k) acc = fmaf(x1[16 * d + k], we[16 * (19 + k) + o], acc);
        edge1[(size_t)e * 16 + o] = acc > 0.f ? acc : 0.f;
    }
#pragma clang loop unroll(disable)
    for (int f = 0; f < 16; ++f) atomicAdd(&acc1[(size_t)d * 16 + f], nv * x1[16 * s + f]);
}

// ---------------------------------------------------------------------------
// conv2 edge + edge_net fused: one wave per FULL 16-edge tile.
// ---------------------------------------------------------------------------
#define S2   49
#define ST   17

__global__ void __launch_bounds__(256) k_conv2_edge(
    const float* __restrict__ x2, const float* __restrict__ edge1,
    const int* __restrict__ ei, const float* __restrict__ we,
    const float* __restrict__ be, const float* __restrict__ dis,
    const float* __restrict__ wen1, const float* __restrict__ ben1,
    const float* __restrict__ wen2, const float* __restrict__ ben2,
    float* __restrict__ acc2, float* __restrict__ out_edge, int E, int Efull) {
    __shared__ float As[8][16 * S2];
    __shared__ float Ts[8][16 * ST];
    __shared__ float nrm[8][16];
    __shared__ int   sdst[8][16];

    const int wv = threadIdx.x >> 5, ln = threadIdx.x & 31;
    const int base = (blockIdx.x * 8 + wv) * 16;
    if (base >= Efull) return;
    float* A = As[wv];
    float* T = Ts[wv];

    __builtin_prefetch(ei + base + 2048, 0, 1);
    __builtin_prefetch(ei + E + base + 2048, 0, 1);
    __builtin_prefetch(edge1 + (size_t)(base + 512) * 16, 0, 1);

    if (ln < 16) {
        const int m = ln, e = base + m;
        const int s = ei[e], d = ei[E + e];
        sdst[wv][m] = d;
        nrm[wv][m]  = dis[s] * dis[d];
        const float* xs = x2 + (size_t)s * 16;
        const float* er = edge1 + (size_t)e * 16;
#pragma unroll
        for (int f = 0; f < 16; ++f) {
            A[m * S2 + f]      = xs[f];
            A[m * S2 + 16 + f] = er[f];
        }
    } else {
        const int m = ln - 16, e = base + m;
        const int d = ei[E + e];
        const float* xd = x2 + (size_t)d * 16;
#pragma unroll
        for (int f = 0; f < 16; ++f) A[m * S2 + 32 + f] = xd[f];
    }
    __syncthreads();

    const int n    = (ln < 16) ? ln : ln - 16;
    const int koff = (ln < 16) ? 0 : 2;
    const int row0 = (ln < 16) ? 0 : 8;

    v2f bf[12];
#pragma unroll
    for (int j = 0; j < 12; ++j) {
        const int k0 = j * 4 + koff;
        bf[j].x = we[k0 * 16 + n];
        bf[j].y = we[(k0 + 1) * 16 + n];
    }
    v8f c; const float bias = be[n];
#pragma unroll
    for (int v = 0; v < 8; ++v) c[v] = bias;
#pragma unroll
    for (int j = 0; j < 12; ++j) {
        v2f a;
        a.x = A[n * S2 + j * 4 + koff];
        a.y = A[n * S2 + j * 4 + koff + 1];
        c = __builtin_amdgcn_wmma_f32_16x16x4_f32(false, a, false, bf[j], (short)0, c,
                                                  false, false);
    }

#pragma unroll
    for (int mm = 0; mm < 8; ++mm) {
        const int m = row0 + mm;
        atomicAdd(&acc2[(size_t)sdst[wv][m] * 16 + n], nrm[wv][m] * A[m * S2 + n]);
    }

#pragma unroll
    for (int v = 0; v < 8; ++v) T[(row0 + v) * ST + n] = c[v] > 0.f ? c[v] : 0.f;
    __syncthreads();

    v8f h; const float b1n = ben1[n];
#pragma unroll
    for (int v = 0; v < 8; ++v) h[v] = b1n;
#pragma unroll
    for (int j = 0; j < 4; ++j) {
        const int k0 = j * 4 + koff;
        v2f a, b;
        a.x = T[n * ST + k0];
        a.y = T[n * ST + k0 + 1];
        b.x = wen1[k0 * 16 + n];
        b.y = wen1[(k0 + 1) * 16 + n];
        h = __builtin_amdgcn_wmma_f32_16x16x4_f32(false, a, false, b, (short)0, h,
                                                  false, false);
    }
    __syncthreads();
#pragma unroll
    for (int v = 0; v < 8; ++v) T[(row0 + v) * ST + n] = h[v] > 0.f ? h[v] : 0.f;
    __syncthreads();

    const int m = n, e = base + m;
    if (ln < 16) {
        float s0 = ben2[0], s1 = ben2[1];
#pragma unroll
        for (int k = 0; k < 16; ++k) {
            const float t = T[m * ST + k];
            s0 = fmaf(t, wen2[3 * k + 0], s0);
            s1 = fmaf(t, wen2[3 * k + 1], s1);
        }
        out_edge[(size_t)e * 3 + 0] = s0;
        out_edge[(size_t)e * 3 + 1] = s1;
    } else {
        float s2 = ben2[2];
#pragma unroll
        for (int k = 0; k < 16; ++k) s2 = fmaf(T[m * ST + k], wen2[3 * k + 2], s2);
        out_edge[(size_t)e * 3 + 2] = s2;
    }
}

// compact scalar tail for (E % 16) edges — not launched when E % 16 == 0
__global__ void k_conv2_tail(const float* __restrict__ x2, const float* __restrict__ edge1,
                             const int* __restrict__ ei, const float* __restrict__ we,
                             const float* __restrict__ be, const float* __restrict__ dis,
                             const float* __restrict__ wen1, const float* __restrict__ ben1,
                             const float* __restrict__ wen2, const float* __restrict__ ben2,
                             float* __restrict__ acc2, float* __restrict__ out_edge,
                             int E, int start) {
    int e = start + blockIdx.x * blockDim.x + threadIdx.x;
    if (e >= E) return;
    const int s = ei[e], d = ei[E + e];
    const float nv = dis[s] * dis[d];
    float e2[16], h[16];
#pragma unroll
    for (int o = 0; o < 16; ++o) {
        float acc = be[o];
#pragma clang loop unroll(disable)
        for (int k = 0; k < 16; ++k) acc = fmaf(x2[16 * s + k], we[16 * k + o], acc);
#pragma clang loop unroll(disable)
        for (int k = 0; k < 16; ++k) acc = fmaf(edge1[(size_t)e * 16 + k], we[16 * (16 + k) + o], acc);
#pragma clang loop unroll(disable)
        for (int k = 0; k < 16; ++k) acc = fmaf(x2[16 * d + k], we[16 * (32 + k) + o], acc);
        e2[o] = acc > 0.f ? acc : 0.f;
    }
#pragma clang loop unroll(disable)
    for (int f = 0; f < 16; ++f) atomicAdd(&acc2[(size_t)d * 16 + f], nv * x2[16 * s + f]);
#pragma unroll
    for (int o = 0; o < 16; ++o) {
        float acc = ben1[o];
#pragma clang loop unroll(disable)
        for (int k = 0; k < 16; ++k) acc = fmaf(e2[k], wen1[16 * k + o], acc);
        h[o] = acc > 0.f ? acc : 0.f;
    }
#pragma unroll
    for (int c = 0; c < 3; ++c) {
        float acc = ben2[c];
#pragma clang loop unroll(disable)
        for (int k = 0; k < 16; ++k) acc = fmaf(h[k], wen2[3 * k + c], acc);
        out_edge[(size_t)e * 3 + c] = acc;
    }
}

// ---------------------------------------------------------------------------
extern "C" void kernel_launch(void* const* d_in, const int* in_sizes, int n_in,
                              void* d_out, int out_size, void* d_ws, size_t ws_size,
                              hipStream_t stream) {
    const float* x     = (const float*)d_in[0];
    const float* eattr = (const float*)d_in[1];
    const int*   ei    = (const int*)d_in[2];
    const float* w_n1  = (const float*)d_in[3];
    const float* b_n1  = (const float*)d_in[4];
    const float* w_e1  = (const float*)d_in[5];
    const float* b_e1  = (const float*)d_in[6];
    const float* w_n2  = (const float*)d_in[7];
    const float* b_n2  = (const float*)d_in[8];
    const float* w_e2  = (const float*)d_in[9];
    const float* b_e2  = (const float*)d_in[10];
    const float* w_nn1 = (const float*)d_in[11];
    const float* b_nn1 = (const float*)d_in[12];
    const float* w_nn2 = (const float*)d_in[13];
    const float* b_nn2 = (const float*)d_in[14];
    const float* w_en1 = (const float*)d_in[15];
    const float* b_en1 = (const float*)d_in[16];
    const float* w_en2 = (const float*)d_in[17];
    const float* b_en2 = (const float*)d_in[18];

    const int NN = in_sizes[0] / 3;              // 100000 nodes
    const int E  = in_sizes[1] / 3;              // 3200000 edges

    float* ws    = (float*)d_ws;
    float* deg   = ws;                           // [N]     -> deg^-0.5
    float* x1    = ws + (size_t)NN;              // [N,16]
    float* acc1  = ws + (size_t)17 * NN;         // [N,16]
    float* x2    = ws + (size_t)33 * NN;         // [N,16]
    float* acc2  = ws + (size_t)49 * NN;         // [N,16]
    float* edge1 = ws + (size_t)65 * NN;         // [E,16]

    float* out_node = (float*)d_out;             // [N,3]
    float* out_edge = (float*)d_out + (size_t)3 * NN;  // [E,3]

    hipMemsetAsync(ws, 0, (size_t)65 * NN * sizeof(float), stream);

    const int T = 256;
    k_deg<<<(E + T - 1) / T, T, 0, stream>>>(ei, deg, E);
    k_dinv<<<(NN + T - 1) / T, T, 0, stream>>>(deg, NN);

    const int ntile  = NN / 16;                  // 6250 full node tiles
    const int Nfull  = ntile * 16;
    const int nrem   = NN - Nfull;               // 0 for this workload
    const int nblk_n = (ntile + 7) / 8;

    const int tiles  = E / 16;                   // 200000 full edge tiles
    const int Efull  = tiles * 16;
    const int erem   = E - Efull;                // 0 for this workload
    const int nblk_e = (tiles + 7) / 8;

    k_lin1_wmma<<<nblk_n, T, 0, stream>>>(x, w_n1, b_n1, x1, Nfull);
    if (nrem > 0)
        k_lin1_tail<<<1, 32, 0, stream>>>(x, w_n1, b_n1, x1, NN, Nfull);

    k_conv1_edge<<<nblk_e, T, 0, stream>>>(x1, eattr, ei, w_e1, b_e1, deg,
                                           edge1, acc1, E, Efull);
    if (erem > 0)
        k_conv1_tail<<<1, 32, 0, stream>>>(x1, eattr, ei, w_e1, b_e1, deg,
                                           edge1, acc1, E, Efull);

    k_lin16_relu_wmma<<<nblk_n, T, 0, stream>>>(acc1, w_n2, b_n2, x2, Nfull);
    if (nrem > 0)
        k_lin16_tail<<<1, 32, 0, stream>>>(acc1, w_n2, b_n2, x2, NN, Nfull);

    k_conv2_edge<<<nblk_e, T, 0, stream>>>(x2, edge1, ei, w_e2, b_e2, deg,
                                           w_en1, b_en1, w_en2, b_en2,
                                           acc2, out_edge, E, Efull);
    if (erem > 0)
        k_conv2_tail<<<1, 32, 0, stream>>>(x2, edge1, ei, w_e2, b_e2, deg,
                                           w_en1, b_en1, w_en2, b_en2,
                                           acc2, out_edge, E, Efull);

    k_node_final_wmma<<<nblk_n, T, 0, stream>>>(acc2, w_nn1, b_nn1, w_nn2, b_nn2,
                                                out_node, Nfull);
    if (nrem > 0)
        k_node_final_tail<<<1, 32, 0, stream>>>(acc2, w_nn1, b_nn1, w_nn2, b_nn2,
                                                out_node, NN, Nfull);
}